// DenseDilatedKnnGraph_63428077027670
// MI455X (gfx1250) — compile-verified
//
#include <hip/hip_runtime.h>
#include <hip/hip_bf16.h>

typedef float v2f __attribute__((ext_vector_type(2)));
typedef float v8f __attribute__((ext_vector_type(8)));

#define BB 2
#define NN 4096
#define CXD 192
#define CCD 64
#define RB (NN / 16)          // 256 row-tiles per batch
#define CHUNK 256             // columns per LDS top-k chunk
#define LDF_STRIDE (CHUNK + 4)
#define KTOT 18
#define KOUT 9
#define EDGE_ELEMS (2 * BB * NN * KOUT)

// ---------------------------------------------------------------------------
// Kernel 1: channel-wise L2 normalize, transpose to row-major (N,C), and
// record per-row self-dot (matches reference's aa/bb computed on normalized
// values in f32).
// ---------------------------------------------------------------------------
__global__ void nrm_kernel(const float* __restrict__ src, float* __restrict__ dstT,
                           float* __restrict__ selfdot, int C) {
    int idx = blockIdx.x * blockDim.x + threadIdx.x; // b*NN + n
    if (idx >= BB * NN) return;
    int b = idx / NN, n = idx % NN;
    const float* p = src + (size_t)b * C * NN + n;
    float ss = 0.f;
    for (int c = 0; c < C; ++c) { float v = p[(size_t)c * NN]; ss += v * v; }
    float nm = fmaxf(sqrtf(ss), 1e-12f);
    float inv = 1.0f / nm;
    float* q = dstT + (size_t)idx * C;
    float sd = 0.f;
    for (int c = 0; c < C; ++c) { float v = p[(size_t)c * NN] * inv; q[c] = v; sd += v * v; }
    selfdot[idx] = sd;
}

// ---------------------------------------------------------------------------
// Kernel 2: fused pass. One block = 16 rows of one batch. 4 waves sweep all
// 4096 columns in 16x16 WMMA tiles (f32 16x16x4). Per tile we derive
// dist_x / dist_code / dist_fused, accumulate the three loss streams,
// per-row argmin/argmax (diag masked), per-(block,column) loss column sums,
// and stream fused distances through LDS for a per-row top-18 selection.
// ---------------------------------------------------------------------------
__global__ __launch_bounds__(128) void pass1_kernel(
    const float* __restrict__ XT, const float* __restrict__ CT,
    const float* __restrict__ SX, const float* __restrict__ SC,
    int* __restrict__ CLO, int* __restrict__ FAR_,
    float* __restrict__ COLK, float* __restrict__ COLR,
    float* __restrict__ INTRA, float* __restrict__ edge_out) {
    __shared__ float lAx[16 * CXD];
    __shared__ float lAc[16 * CCD];
    __shared__ float lDf[16 * LDF_STRIDE];
    __shared__ float rMinV[4][16];
    __shared__ int rMinI[4][16];
    __shared__ float rMaxV[4][16];
    __shared__ int rMaxI[4][16];
    __shared__ float wIntra[4];

    int blk = blockIdx.x;
    int b = blk / RB;
    int ibl = blk % RB;
    int ibase = ibl * 16;
    int tid = threadIdx.x;
    int wv = tid >> 5;
    int lane = tid & 31;
    int nloc = lane & 15;   // column within tile for C/D fragment
    int hi = lane >> 4;     // half-wave select
    int mbase = hi * 8;     // rows owned by this lane in C/D fragment

    // Stage the 16-row A tiles (x and code) into LDS, coalesced.
    for (int t = tid; t < 16 * CXD; t += 128)
        lAx[t] = XT[((size_t)(b * NN + ibase + t / CXD)) * CXD + (t % CXD)];
    for (int t = tid; t < 16 * CCD; t += 128)
        lAc[t] = CT[((size_t)(b * NN + ibase + t / CCD)) * CCD + (t % CCD)];
    __syncthreads();

    float sxi[8], sci[8];
#pragma unroll
    for (int v = 0; v < 8; ++v) {
        sxi[v] = SX[b * NN + ibase + mbase + v];
        sci[v] = SC[b * NN + ibase + mbase + v];
    }
    float minV[8], maxV[8];
    int minI[8], maxI[8];
#pragma unroll
    for (int v = 0; v < 8; ++v) {
        minV[v] = 3.0e38f; minI[v] = 0x7fffffff;
        maxV[v] = -3.0e38f; maxI[v] = 0x7fffffff;
    }
    float intra = 0.f;

    // Streaming top-18 state (live in selector threads tid<16).
    float tv[KTOT];
    int ti[KTOT];
#pragma unroll
    for (int k = 0; k < KTOT; ++k) { tv[k] = 3.0e38f; ti[k] = 0x7fffffff; }

    for (int ch = 0; ch < NN / CHUNK; ++ch) {
        int cbase = ch * CHUNK;
        for (int t = wv; t < CHUNK / 16; t += 4) {
            int jbase = cbase + t * 16;
            int jg = jbase + nloc;
            const float* Bx = &XT[((size_t)(b * NN + jg)) * CXD];
            const float* Bc = &CT[((size_t)(b * NN + jg)) * CCD];
            // Prefetch the row this lane will need 4 tiles (64 columns) ahead.
            __builtin_prefetch(&XT[((size_t)(b * NN + ((jg + 64) & (NN - 1)))) * CXD], 0, 1);

            v8f ax = {0, 0, 0, 0, 0, 0, 0, 0};
            v8f ac = {0, 0, 0, 0, 0, 0, 0, 0};
#pragma unroll 4
            for (int kk = 0; kk < CXD; kk += 4) {
                v2f a = *(const v2f*)&lAx[nloc * CXD + kk + 2 * hi];
                v2f bq = *(const v2f*)&Bx[kk + 2 * hi];
                ax = __builtin_amdgcn_wmma_f32_16x16x4_f32(false, a, false, bq,
                                                           (short)0, ax, false, false);
            }
#pragma unroll 4
            for (int kk = 0; kk < CCD; kk += 4) {
                v2f a = *(const v2f*)&lAc[nloc * CCD + kk + 2 * hi];
                v2f bq = *(const v2f*)&Bc[kk + 2 * hi];
                ac = __builtin_amdgcn_wmma_f32_16x16x4_f32(false, a, false, bq,
                                                           (short)0, ac, false, false);
            }
            float sxj = SX[b * NN + jg];
            float scj = SC[b * NN + jg];
            float colK = 0.f, colR = 0.f;
#pragma unroll
            for (int v = 0; v < 8; ++v) {
                float rawx = sxi[v] + sxj - 2.0f * ax[v];
                float rawc = sci[v] + scj - 2.0f * ac[v];
                float dx = fmaxf(rawx, 0.f);
                float dc = fmaxf(rawc, 0.f);
                float dce = fmaxf(dc, 1e-5f);
                intra += -dce * (dx - 0.12f);
                colK += -dce * (dx - 0.2f);
                colR += -dce * (dx - 1.0f);
                int ig = ibase + mbase + v;
                float dsel = (ig == jg) ? 3.0e38f : dx;
                if (dsel < minV[v] || (dsel == minV[v] && jg < minI[v])) { minV[v] = dsel; minI[v] = jg; }
                float dmx = (ig == jg) ? -3.0e38f : dx;
                if (dmx > maxV[v] || (dmx == maxV[v] && jg < maxI[v])) { maxV[v] = dmx; maxI[v] = jg; }
                lDf[(mbase + v) * LDF_STRIDE + t * 16 + nloc] = fmaxf(rawx + rawc, 0.f);
            }
            // Fold the two half-wave row groups into full column sums; one
            // store per column per block (no atomics -> deterministic).
            colK += __shfl_xor(colK, 16);
            colR += __shfl_xor(colR, 16);
            if (hi == 0) {
                COLK[((size_t)(b * RB + ibl)) * NN + jg] = colK;
                COLR[((size_t)(b * RB + ibl)) * NN + jg] = colR;
            }
        }
        __syncthreads();
        // Per-row top-18 merge of this 256-column chunk (stable: (dist,idx)).
        if (tid < 16) {
            const float* row = &lDf[tid * LDF_STRIDE];
            for (int c = 0; c < CHUNK; ++c) {
                float v = row[c];
                int j = cbase + c;
                if (v < tv[KTOT - 1] || (v == tv[KTOT - 1] && j < ti[KTOT - 1])) {
                    int p = KTOT - 1;
                    while (p > 0 && (v < tv[p - 1] || (v == tv[p - 1] && j < ti[p - 1]))) {
                        tv[p] = tv[p - 1]; ti[p] = ti[p - 1]; --p;
                    }
                    tv[p] = v; ti[p] = j;
                }
            }
        }
        __syncthreads();
    }

    // Cross-lane argmin/argmax within each half-wave (same row set).
#pragma unroll
    for (int off = 1; off <= 8; off <<= 1) {
#pragma unroll
        for (int v = 0; v < 8; ++v) {
            float ov = __shfl_xor(minV[v], off);
            int oi = __shfl_xor(minI[v], off);
            if (ov < minV[v] || (ov == minV[v] && oi < minI[v])) { minV[v] = ov; minI[v] = oi; }
            float xv = __shfl_xor(maxV[v], off);
            int xi = __shfl_xor(maxI[v], off);
            if (xv > maxV[v] || (xv == maxV[v] && xi < maxI[v])) { maxV[v] = xv; maxI[v] = xi; }
        }
    }
    if (nloc == 0) {
#pragma unroll
        for (int v = 0; v < 8; ++v) {
            rMinV[wv][mbase + v] = minV[v]; rMinI[wv][mbase + v] = minI[v];
            rMaxV[wv][mbase + v] = maxV[v]; rMaxI[wv][mbase + v] = maxI[v];
        }
    }
#pragma unroll
    for (int off = 1; off < 32; off <<= 1) intra += __shfl_xor(intra, off);
    if (lane == 0) wIntra[wv] = intra;
    __syncthreads();

    if (tid < 16) {
        float bv = rMinV[0][tid]; int bi = rMinI[0][tid];
        for (int w = 1; w < 4; ++w) {
            float ov = rMinV[w][tid]; int oi = rMinI[w][tid];
            if (ov < bv || (ov == bv && oi < bi)) { bv = ov; bi = oi; }
        }
        CLO[b * NN + ibase + tid] = bi;
        float xv = rMaxV[0][tid]; int xi = rMaxI[0][tid];
        for (int w = 1; w < 4; ++w) {
            float ov = rMaxV[w][tid]; int oi = rMaxI[w][tid];
            if (ov > xv || (ov == xv && oi < xi)) { xv = ov; xi = oi; }
        }
        FAR_[b * NN + ibase + tid] = xi;
        // edge_index output, dilation 2: take k = 0,2,...,16
        int i = ibase + tid;
        for (int k = 0; k < KOUT; ++k) {
            edge_out[((size_t)(b * NN + i)) * KOUT + k] = (float)ti[2 * k];
            edge_out[(size_t)BB * NN * KOUT + ((size_t)(b * NN + i)) * KOUT + k] = (float)i;
        }
    }
    if (tid == 0) INTRA[blk] = ((wIntra[0] + wIntra[1]) + wIntra[2]) + wIntra[3];
}

// ---------------------------------------------------------------------------
// Kernel 3: deterministically reduce per-row-block column sums over the 256
// row blocks -> total column sums per (b, j).
// ---------------------------------------------------------------------------
__global__ void colred_kernel(const float* __restrict__ COLK, const float* __restrict__ COLR,
                              float* __restrict__ KT, float* __restrict__ RT) {
    int idx = blockIdx.x * blockDim.x + threadIdx.x; // b*NN + j
    if (idx >= BB * NN) return;
    int b = idx / NN, j = idx % NN;
    float sk = 0.f, sr = 0.f;
    for (int rb = 0; rb < RB; ++rb) {
        sk += COLK[((size_t)(b * RB + rb)) * NN + j];
        sr += COLR[((size_t)(b * RB + rb)) * NN + j];
    }
    KT[idx] = sk;
    RT[idx] = sr;
}

// ---------------------------------------------------------------------------
// Kernel 4: single-block fixed-order reduction. Gathers column sums through
// closest/farthest (mean over j of colsum[cl(j)] == full knn/rand mean) and
// combines the three weighted loss terms into drl.
// ---------------------------------------------------------------------------
__global__ void final_kernel(const float* __restrict__ INTRA, const int* __restrict__ CLO,
                             const int* __restrict__ FAR_, const float* __restrict__ KT,
                             const float* __restrict__ RT, float* __restrict__ out) {
    __shared__ float rA[256], rK[256], rR[256];
    int tid = threadIdx.x;
    float sA = 0.f, sK = 0.f, sR = 0.f;
    for (int i = tid; i < BB * RB; i += 256) sA += INTRA[i];
    for (int i = tid; i < BB * NN; i += 256) {
        int b = i / NN;
        sK += KT[b * NN + CLO[i]];
        sR += RT[b * NN + FAR_[i]];
    }
    rA[tid] = sA; rK[tid] = sK; rR[tid] = sR;
    __syncthreads();
    for (int s = 128; s > 0; s >>= 1) {
        if (tid < s) { rA[tid] += rA[tid + s]; rK[tid] += rK[tid + s]; rR[tid] += rR[tid + s]; }
        __syncthreads();
    }
    if (tid == 0) {
        float scale = (float)BB * (float)NN * (float)NN;
        out[EDGE_ELEMS] = 0.1f * (rA[0] / scale) + 1.0f * (rK[0] / scale) + 0.15f * (rR[0] / scale);
    }
}

// ---------------------------------------------------------------------------
extern "C" void kernel_launch(void* const* d_in, const int* in_sizes, int n_in,
                              void* d_out, int out_size, void* d_ws, size_t ws_size,
                              hipStream_t stream) {
    (void)in_sizes; (void)n_in; (void)out_size; (void)ws_size;
    const float* x = (const float*)d_in[0];     // (2,192,4096,1) f32
    const float* code = (const float*)d_in[1];  // (2,64,4096,1)  f32
    float* out = (float*)d_out;                 // edge_index (as float) + drl

    char* w = (char*)d_ws;
    size_t off = 0;
    auto carve = [&](size_t bytes) -> char* {
        char* p = w + off;
        off = (off + bytes + 255) & ~(size_t)255;
        return p;
    };
    float* XT = (float*)carve(sizeof(float) * (size_t)BB * NN * CXD);
    float* CT = (float*)carve(sizeof(float) * (size_t)BB * NN * CCD);
    float* SX = (float*)carve(sizeof(float) * (size_t)BB * NN);
    float* SC = (float*)carve(sizeof(float) * (size_t)BB * NN);
    int* CLO = (int*)carve(sizeof(int) * (size_t)BB * NN);
    int* FAR_ = (int*)carve(sizeof(int) * (size_t)BB * NN);
    float* COLK = (float*)carve(sizeof(float) * (size_t)BB * RB * NN);
    float* COLR = (float*)carve(sizeof(float) * (size_t)BB * RB * NN);
    float* KT = (float*)carve(sizeof(float) * (size_t)BB * NN);
    float* RT = (float*)carve(sizeof(float) * (size_t)BB * NN);
    float* INTRA = (float*)carve(sizeof(float) * (size_t)BB * RB);

    nrm_kernel<<<(BB * NN + 255) / 256, 256, 0, stream>>>(x, XT, SX, CXD);
    nrm_kernel<<<(BB * NN + 255) / 256, 256, 0, stream>>>(code, CT, SC, CCD);
    pass1_kernel<<<BB * RB, 128, 0, stream>>>(XT, CT, SX, SC, CLO, FAR_, COLK, COLR, INTRA, out);
    colred_kernel<<<(BB * NN + 255) / 256, 256, 0, stream>>>(COLK, COLR, KT, RT);
    final_kernel<<<1, 256, 0, stream>>>(INTRA, CLO, FAR_, KT, RT, out);
}